// MonoEncoder_63857573757445
// MI455X (gfx1250) — compile-verified
//
#include <hip/hip_runtime.h>

#define D_FEAT     32
#define EPS_NN     1e-6f
#define NEG_SLOPE  0.01f

typedef __attribute__((ext_vector_type(2))) float v2f;
typedef __attribute__((ext_vector_type(8))) float v8f;

// ---------------------------------------------------------------- deg = 1.0
__global__ void __launch_bounds__(256) k_init_deg(float* __restrict__ deg, int n) {
    int i = blockIdx.x * blockDim.x + threadIdx.x;
    if (i < n) deg[i] = 1.0f;
}

// ------------------------------------------------- deg[dst] += 1 per edge
__global__ void __launch_bounds__(256) k_count_deg(const int* __restrict__ dst,
                                                   float* __restrict__ deg, int e) {
    int i = blockIdx.x * blockDim.x + threadIdx.x;
    if (i < e) atomicAdd(&deg[dst[i]], 1.0f);
}

// ---------------------------------------------------------------- h = x @ W
// One wave computes a 16x32 row-tile via V_WMMA_F32_16X16X4_F32.
// A (16x4 f32): lanes 0-15 hold M=0..15 with VGPR0=K0,VGPR1=K1; lanes 16-31 K2,K3.
// B (4x16 f32): VGPR r, lanes 0-15 -> K=r, lanes 16-31 -> K=r+2; N=lane&15.
// C/D (16x16):  VGPR r, lanes 0-15 -> M=r, lanes 16-31 -> M=r+8; N=lane&15.
__global__ void __launch_bounds__(256) k_gemm_wmma(const float* __restrict__ x,
                                                   const float* __restrict__ W,
                                                   float* __restrict__ h,
                                                   int nRowTiles) {
    int wave = (blockIdx.x * blockDim.x + threadIdx.x) >> 5;   // wave-uniform
    int lane = threadIdx.x & 31;
    if (wave >= nRowTiles) return;                             // whole wave exits: EXEC stays all-1s for WMMA

    const int row0  = wave * 16;
    const int m     = lane & 15;       // A row / B,C column
    const int khalf = lane >> 4;       // 0: K{0,1} / M 0..7   1: K{2,3} / M 8..15

    v8f acc0 = {};   // columns 0..15
    v8f acc1 = {};   // columns 16..31

    #pragma unroll
    for (int k = 0; k < 8; ++k) {                 // K = 32 in steps of 4
        const int kb = k * 4 + khalf * 2;
        v2f a = *(const v2f*)(x + (row0 + m) * D_FEAT + kb);

        v2f b0, b1;                                // B fragments (two 4x16 col-tiles)
        b0.x = W[(kb + 0) * D_FEAT + m];
        b0.y = W[(kb + 1) * D_FEAT + m];
        b1.x = W[(kb + 0) * D_FEAT + 16 + m];
        b1.y = W[(kb + 1) * D_FEAT + 16 + m];

        acc0 = __builtin_amdgcn_wmma_f32_16x16x4_f32(false, a, false, b0,
                                                     (short)0, acc0, false, false);
        acc1 = __builtin_amdgcn_wmma_f32_16x16x4_f32(false, a, false, b1,
                                                     (short)0, acc1, false, false);
    }

    #pragma unroll
    for (int r = 0; r < 8; ++r) {
        const int mo = r + khalf * 8;
        h[(row0 + mo) * D_FEAT + m]      = acc0[r];
        h[(row0 + mo) * D_FEAT + 16 + m] = acc1[r];
    }
}

// --------------------------------- out = h / deg  (self-loop: dis[i]^2 = 1/deg[i])
// 8 threads per node, float4 each. Writes every element of d_out (clears poison).
__global__ void __launch_bounds__(256) k_selfloop(const float* __restrict__ h,
                                                  const float* __restrict__ deg,
                                                  float* __restrict__ out, int n) {
    int t    = blockIdx.x * blockDim.x + threadIdx.x;
    int node = t >> 3;
    int c    = (t & 7) * 4;
    if (node < n) {
        float inv = 1.0f / deg[node];
        float4 hv = *(const float4*)(h + node * D_FEAT + c);
        float4 o  = make_float4(hv.x * inv, hv.y * inv, hv.z * inv, hv.w * inv);
        *(float4*)(out + node * D_FEAT + c) = o;
    }
}

// ------------------------- out[dst] += h[src] * rsqrt(deg[src])*rsqrt(deg[dst])
// 8 threads per edge, float4 gather + 4 global f32 atomics each.
__global__ void __launch_bounds__(256) k_edges(const int* __restrict__ src,
                                               const int* __restrict__ dst,
                                               const float* __restrict__ h,
                                               const float* __restrict__ deg,
                                               float* __restrict__ out, int e) {
    int t    = blockIdx.x * blockDim.x + threadIdx.x;
    int edge = t >> 3;
    int c    = (t & 7) * 4;
    if (edge < e) {
        int s = src[edge];
        int d = dst[edge];
        float norm = __frsqrt_rn(deg[s]) * __frsqrt_rn(deg[d]);
        float4 hv  = *(const float4*)(h + s * D_FEAT + c);
        float* o   = out + d * D_FEAT + c;
        atomicAdd(o + 0, hv.x * norm);
        atomicAdd(o + 1, hv.y * norm);
        atomicAdd(o + 2, hv.z * norm);
        atomicAdd(o + 3, hv.w * norm);
    }
}

// ------------------- NodeNorm (per-row mean/biased-var) + LeakyReLU, in place
// One wave32 per node row; feature dim == 32 == warpSize.
__global__ void __launch_bounds__(256) k_finalize(float* __restrict__ out,
                                                  const float* __restrict__ bias,
                                                  int n) {
    int row  = (blockIdx.x * blockDim.x + threadIdx.x) >> 5;
    int lane = threadIdx.x & 31;
    if (row >= n) return;

    float v  = out[row * D_FEAT + lane] + bias[lane];
    float s  = v;
    float sq = v * v;
    #pragma unroll
    for (int off = 16; off > 0; off >>= 1) {
        s  += __shfl_xor(s,  off, 32);
        sq += __shfl_xor(sq, off, 32);
    }
    float mean = s  * (1.0f / 32.0f);
    float var  = sq * (1.0f / 32.0f) - mean * mean;
    float inv  = __frsqrt_rn(var + EPS_NN);
    float y    = (v - mean) * inv;
    out[row * D_FEAT + lane] = (y >= 0.0f) ? y : NEG_SLOPE * y;
}

extern "C" void kernel_launch(void* const* d_in, const int* in_sizes, int n_in,
                              void* d_out, int out_size, void* d_ws, size_t ws_size,
                              hipStream_t stream) {
    const float* x  = (const float*)d_in[0];
    const int*   ei = (const int*)  d_in[1];
    const float* W  = (const float*)d_in[2];
    const float* b  = (const float*)d_in[3];

    const int n = in_sizes[0] / D_FEAT;   // 100000
    const int e = in_sizes[1] / 2;        // 1600000
    const int* src = ei;                  // edge_index[0]
    const int* dst = ei + e;              // edge_index[1]

    float* out = (float*)d_out;
    float* deg = (float*)d_ws;            // [n]
    float* h   = deg + n;                 // [n*32]

    const int B = 256;

    // 1) deg = 1.0
    k_init_deg<<<(n + B - 1) / B, B, 0, stream>>>(deg, n);
    // 2) deg[dst] += 1
    k_count_deg<<<(e + B - 1) / B, B, 0, stream>>>(dst, deg, e);
    // 3) h = x @ W  (WMMA f32 16x16x4; 8 waves/block, 1 row-tile/wave)
    const int nRowTiles = n / 16;                       // N divisible by 16
    const int wavesPerBlock = B / 32;
    k_gemm_wmma<<<(nRowTiles + wavesPerBlock - 1) / wavesPerBlock, B, 0, stream>>>(x, W, h, nRowTiles);
    // 4) out = h / deg  (self-loop contribution; fully initializes d_out)
    k_selfloop<<<((n * 8) + B - 1) / B, B, 0, stream>>>(h, deg, out, n);
    // 5) edge scatter-add
    k_edges<<<((e * 8) + B - 1) / B, B, 0, stream>>>(src, dst, h, deg, out, e);
    // 6) NodeNorm + bias + LeakyReLU
    k_finalize<<<((n * 32) + B - 1) / B, B, 0, stream>>>(out, b, n);
}